// Linear_83820581748941
// MI455X (gfx1250) — compile-verified
//
#include <hip/hip_runtime.h>

typedef __attribute__((ext_vector_type(2))) float v2f;
typedef __attribute__((ext_vector_type(4))) float v4f;
typedef __attribute__((ext_vector_type(8))) float v8f;

#define DIMC 148              // 32*1 + 16*3 + 8*5 + 4*7
#define ROWS_PER_BLOCK 64     // 4 waves x 16 rows
#define THREADS 128
#define MAX_BLOCKS 2048       // grid-stride over tiles; ~4 tiles/block @ N=524288

// ---------------------------------------------------------------------------
// B-matrix (weight) fragments for V_WMMA_F32_16X16X4_F32.
// B is K x N = 4 x 16 per WMMA: lane&15 = column v, lane>>4 selects the K pair:
// b.x = W[kb][v], b.y = W[kb+1][v]. Scale (mul^-0.5) folded in. v >= mul -> 0.
// ---------------------------------------------------------------------------
template <int MUL>
__device__ inline void load_bfrag(const float* __restrict__ w, float scale,
                                  v2f* bfrag, int lane) {
  constexpr int NT = (MUL + 15) / 16;
  constexpr int KS = MUL / 4;
#pragma unroll
  for (int nt = 0; nt < NT; ++nt) {
    const int v = nt * 16 + (lane & 15);
#pragma unroll
    for (int ks = 0; ks < KS; ++ks) {
      const int kb = ks * 4 + ((lane >> 4) << 1);
      v2f b;
      b.x = 0.0f;
      b.y = 0.0f;
      if (MUL >= 16 || v < MUL) {  // constant-folds for MUL >= 16
        b.x = w[kb * MUL + v] * scale;
        b.y = w[(kb + 1) * MUL + v] * scale;
      }
      bfrag[nt * KS + ks] = b;
    }
  }
}

// ---------------------------------------------------------------------------
// One irrep on one wave's 16-row LDS tile, in place.
// Virtual GEMM row m = n_local*D + i; K = N = MUL.
// A (16x4 f32): lane&15 = M, lane>>4 selects K pair within the 4-wide chunk.
// D: VGPR r -> M = r + (lane>>4)*8, col = lane&15.
// Scatter offsets: mm = mt*16 + r + h*8 with mt,r unroll-constants and
// h = lane>>4 a single bit -> both (mm/D)*DIMC + mm%D candidates are
// compile-time constants, selected with one cndmask (no runtime division).
// ---------------------------------------------------------------------------
template <int MUL, int D, int OFF>
__device__ inline void irrep_compute(float* __restrict__ xw, const v2f* bfrag,
                                     int lane) {
  constexpr int NT = (MUL + 15) / 16;
  constexpr int KS = MUL / 4;
  const int h = lane >> 4;
#pragma unroll
  for (int mt = 0; mt < D; ++mt) {  // M tiles: 16*D virtual rows total
    const int m = mt * 16 + (lane & 15);
    const int nl = m / D;           // lane-dependent; loop-invariant -> hoisted
    const int i = m - nl * D;
    const float* ap = xw + nl * DIMC + OFF + i;

    v2f a[KS];
#pragma unroll
    for (int ks = 0; ks < KS; ++ks) {
      const int kb = ks * 4 + (h << 1);
      a[ks].x = ap[(kb + 0) * D];
      a[ks].y = ap[(kb + 1) * D];
    }

    v8f dacc[NT];
#pragma unroll
    for (int nt = 0; nt < NT; ++nt) {
      v8f c = {};
#pragma unroll
      for (int ks = 0; ks < KS; ++ks) {
        c = __builtin_amdgcn_wmma_f32_16x16x4_f32(
            /*neg_a=*/false, a[ks], /*neg_b=*/false, bfrag[nt * KS + ks],
            /*c_mod=*/(short)0, c, /*reuse_a=*/false, /*reuse_b=*/false);
      }
      dacc[nt] = c;
    }

#pragma unroll
    for (int nt = 0; nt < NT; ++nt) {
      const int v = nt * 16 + (lane & 15);
      if (MUL >= 16 || v < MUL) {
        float* sp = xw + OFF + v * D;
#pragma unroll
        for (int r = 0; r < 8; ++r) {
          const int mm0 = mt * 16 + r;                  // compile-time
          const int mm1 = mm0 + 8;                      // compile-time
          const int k0 = (mm0 / D) * DIMC + (mm0 % D);  // compile-time
          const int k1 = (mm1 / D) * DIMC + (mm1 % D);  // compile-time
          sp[h ? k1 : k0] = dacc[nt][r];                // cndmask + ds_store
        }
      }
    }
  }
}

__global__ __launch_bounds__(THREADS) void eq_linear_wmma(
    const float* __restrict__ x, const float* __restrict__ w0,
    const float* __restrict__ w1, const float* __restrict__ w2,
    const float* __restrict__ w3, float* __restrict__ y, long long nrows) {
  __shared__ float lds[ROWS_PER_BLOCK * DIMC];  // 37,888 B

  const int tid = threadIdx.x;
  const int lane = tid & 31;
  const int wave = tid >> 5;

  // Weight fragments resident in VGPRs for the whole kernel.
  v2f b0[16], b1[4], b2[2], b3[1];
  load_bfrag<32>(w0, 0.17677669529663687f, b0, lane);  // 32^-0.5
  load_bfrag<16>(w1, 0.25f, b1, lane);                 // 16^-0.5
  load_bfrag<8>(w2, 0.35355339059327379f, b2, lane);   // 8^-0.5
  load_bfrag<4>(w3, 0.5f, b3, lane);                   // 4^-0.5

  constexpr int NVEC = ROWS_PER_BLOCK * DIMC / 4;  // 2368 float4 per tile
  constexpr int ITER = NVEC / THREADS;             // 18 full rounds
  constexpr int REM = NVEC % THREADS;              // 64 tail elements
  const long long totalVec = nrows * (DIMC / 4);
  const long long tiles = (nrows + ROWS_PER_BLOCK - 1) / ROWS_PER_BLOCK;
  const v4f* __restrict__ gx = (const v4f*)x;
  v4f* __restrict__ gy = (v4f*)y;
  v4f* lx = (v4f*)lds;
  float* xw = lds + wave * 16 * DIMC;

  // Grid-stride over 64-row tiles: all LDS addressing above is loop-invariant.
  for (long long t = blockIdx.x; t < tiles; t += gridDim.x) {
    const long long vecBase = t * (long long)NVEC;
    const bool full = (vecBase + NVEC) <= totalVec;  // always true @ N=524288

    // -------- global -> LDS (coalesced b128, non-temporal, high MLP) --------
    if (full) {
      const v4f* __restrict__ gsrc = gx + vecBase;
#pragma unroll 9
      for (int k = 0; k < ITER; ++k) {
        const int iv = tid + k * THREADS;
        lx[iv] = __builtin_nontemporal_load(&gsrc[iv]);
      }
      if (tid < REM) {
        const int iv = tid + ITER * THREADS;
        lx[iv] = __builtin_nontemporal_load(&gsrc[iv]);
      }
    } else {  // partial tail tile (not taken for N=524288)
      for (int iv = tid; iv < NVEC; iv += THREADS) {
        const long long g = vecBase + iv;
        if (g < totalVec) lx[iv] = gx[g];
      }
    }
    __syncthreads();

    // Each wave owns 16 rows; all irreps via WMMA, in place.
    irrep_compute<32, 1, 0>(xw, b0, lane);    // 16 WMMAs
    irrep_compute<16, 3, 32>(xw, b1, lane);   // 12 WMMAs
    irrep_compute<8, 5, 80>(xw, b2, lane);    // 10 WMMAs
    irrep_compute<4, 7, 120>(xw, b3, lane);   //  7 WMMAs
    __syncthreads();

    // -------- LDS -> global (coalesced b128, non-temporal streaming) --------
    if (full) {
      v4f* __restrict__ gdst = gy + vecBase;
#pragma unroll 9
      for (int k = 0; k < ITER; ++k) {
        const int iv = tid + k * THREADS;
        __builtin_nontemporal_store(lx[iv], &gdst[iv]);
      }
      if (tid < REM) {
        const int iv = tid + ITER * THREADS;
        __builtin_nontemporal_store(lx[iv], &gdst[iv]);
      }
    } else {
      for (int iv = tid; iv < NVEC; iv += THREADS) {
        const long long g = vecBase + iv;
        if (g < totalVec) gy[g] = lx[iv];
      }
    }
    __syncthreads();  // protect LDS before next iteration's load
  }
}

extern "C" void kernel_launch(void* const* d_in, const int* in_sizes, int n_in,
                              void* d_out, int out_size, void* d_ws,
                              size_t ws_size, hipStream_t stream) {
  (void)n_in;
  (void)out_size;
  (void)d_ws;
  (void)ws_size;
  const float* x = (const float*)d_in[0];
  const float* w0 = (const float*)d_in[1];
  const float* w1 = (const float*)d_in[2];
  const float* w2 = (const float*)d_in[3];
  const float* w3 = (const float*)d_in[4];
  float* y = (float*)d_out;

  const long long nrows = (long long)in_sizes[0] / DIMC;  // 524288
  const long long tiles = (nrows + ROWS_PER_BLOCK - 1) / ROWS_PER_BLOCK;
  const unsigned blocks =
      (unsigned)(tiles < (long long)MAX_BLOCKS ? tiles : (long long)MAX_BLOCKS);
  eq_linear_wmma<<<dim3(blocks), dim3(THREADS), 0, stream>>>(x, w0, w1, w2, w3,
                                                             y, nrows);
}